// DGBasedGaussianKLD_34299608826308
// MI455X (gfx1250) — compile-verified
//
#include <hip/hip_runtime.h>
#include <math.h>

typedef __attribute__((ext_vector_type(2))) float v2f;
typedef __attribute__((ext_vector_type(8))) float v8f;

#define LOG2PI_F 1.8378770664093453f
#define LOG2E_F  1.4426950408889634f   // log2(e)
#define LN2_F    0.6931471805599453f   // ln(2)

// Problem constants (fixed by setup_inputs)
constexpr int BATCH = 1024;
constexpr int DIMZ  = 32;
constexpr int NS    = 32;
constexpr int AGG   = 256;
constexpr int NC    = BATCH / AGG;        // 4 chunks
constexpr int SAMPLES = AGG * NS;         // 8192 samples per chunk
constexpr int STILES  = SAMPLES / 16;     // 512 sample tiles (16 rows each)
constexpr int JT      = AGG / 16;         // 16 component tiles
constexpr int TPW     = 8;                // sample tiles handled per wave
constexpr int GROUPS  = STILES / TPW;     // 64 wave-groups per (chunk, dim)

// Workspace float offsets
constexpr int OFF_COEFA = 0;
constexpr int OFF_COEFB = OFF_COEFA + BATCH * DIMZ;   // 32768
constexpr int OFF_COEFC = OFF_COEFB + BATCH * DIMZ;   // 65536
constexpr int OFF_STD   = OFF_COEFC + BATCH * DIMZ;   // 98304
constexpr int OFF_LOGQ  = OFF_STD   + BATCH * DIMZ;   // 131072
constexpr int OFF_ZSQ   = OFF_LOGQ  + NC * DIMZ;      // +128

// ---------------------------------------------------------------------------
// Phase 1: per-(batch-row, dim) coefficients for the K=3 contraction, with
// log2(e) pre-folded so the WMMA result is log2(q) and exp becomes a bare
// v_exp_f32 (no scaling mul on the VALU path):
//   log2_q(z) = coefA*z^2 + coefB*z + coefC
// ---------------------------------------------------------------------------
__global__ void kld_precomp(const float* __restrict__ mean,
                            const float* __restrict__ logvar,
                            float* __restrict__ coefA,
                            float* __restrict__ coefB,
                            float* __restrict__ coefC,
                            float* __restrict__ stdv) {
    int i = blockIdx.x * blockDim.x + threadIdx.x;
    if (i >= BATCH * DIMZ) return;
    float lv = logvar[i];
    float m  = mean[i];
    float a  = __expf(-lv);
    coefA[i] = -0.5f * a * LOG2E_F;
    coefB[i] = m * a * LOG2E_F;
    coefC[i] = -0.5f * (m * m * a + lv + LOG2PI_F) * LOG2E_F;
    stdv[i]  = __expf(0.5f * lv);
}

// ---------------------------------------------------------------------------
// Phase 2: per wave, one (chunk c, dim k, group of 8 sample tiles).
// WMMA f32 16x16x4 computes 16-sample x 16-component log2_q tiles; two
// independent WMMA/exp/accumulate streams keep the trans pipe saturated.
// ---------------------------------------------------------------------------
__global__ __launch_bounds__(256) void kld_main(const float* __restrict__ mean,
                                                const float* __restrict__ eps,
                                                const float* __restrict__ coefA,
                                                const float* __restrict__ coefB,
                                                const float* __restrict__ coefC,
                                                const float* __restrict__ stdv,
                                                float* __restrict__ logqSum,
                                                float* __restrict__ zsqSum) {
    const int waveInBlk = threadIdx.x >> 5;
    const int lane      = threadIdx.x & 31;
    const int waveId    = blockIdx.x * (blockDim.x >> 5) + waveInBlk;

    const int c   = waveId / (DIMZ * GROUPS);
    const int rem = waveId % (DIMZ * GROUPS);
    const int k   = rem / GROUPS;
    const int g   = rem % GROUPS;

    const bool lo = lane < 16;
    const int  ln = lane & 15;

    // Preload the 16 B tiles (component coefficients for this chunk/dim).
    // B(4x16): lanes 0-15 -> (K0,K1)=(coefA,coefB) for col N=ln;
    //          lanes 16-31 -> (K2,K3)=(coefC,0).
    v2f Bt[JT];
#pragma unroll
    for (int t = 0; t < JT; ++t) {
        const int j   = t * 16 + ln;                 // component within chunk
        const int idx = (c * AGG + j) * DIMZ + k;
        if (lo) { Bt[t].x = coefA[idx]; Bt[t].y = coefB[idx]; }
        else    { Bt[t].x = coefC[idx]; Bt[t].y = 0.0f; }
    }

    float logqAcc = 0.0f;   // Sum over this wave's samples of log(S/256)
    float zsqAcc  = 0.0f;   // Sum over this wave's samples of z^2

    for (int ti = 0; ti < TPW; ++ti) {
        const int tile = g * TPW + ti;               // sample tile in [0,512)
        const int b    = c * AGG + (tile >> 1);      // batch row (16 | 32)
        const int ssb  = (tile & 1) * 16;            // sample-sub index base

        // A(16x4): lanes 0-15 -> (K0,K1)=(z^2, z) for row M=ln;
        //          lanes 16-31 -> (K2,K3)=(1, 0).
        v2f At;
        if (lo) {
            const float mu = mean[b * DIMZ + k];
            const float sd = stdv[b * DIMZ + k];
            const float e  = eps[(b * NS + ssb + ln) * DIMZ + k];
            const float z  = fmaf(e, sd, mu);
            const float z2 = z * z;
            zsqAcc += z2;
            At.x = z2; At.y = z;
        } else {
            At.x = 1.0f; At.y = 0.0f;
        }

        // Two independent accumulation streams (A: even tiles, B: odd tiles)
        // with explicit per-element exp temporaries so the scheduler can keep
        // 8+ v_exp_f32 in flight and use adds (not v_nop) as TRANS spacing.
        float esumA[8] = {0.f, 0.f, 0.f, 0.f, 0.f, 0.f, 0.f, 0.f};
        float esumB[8] = {0.f, 0.f, 0.f, 0.f, 0.f, 0.f, 0.f, 0.f};
#pragma unroll
        for (int t = 0; t < JT; t += 2) {
            v8f cz0 = {};
            v8f cz1 = {};
            // D[m][n] = z2_m*coefA_n + z_m*coefB_n + coefC_n = log2(q_mn)
            v8f d0 = __builtin_amdgcn_wmma_f32_16x16x4_f32(
                false, At, false, Bt[t],     (short)0, cz0, false, false);
            v8f d1 = __builtin_amdgcn_wmma_f32_16x16x4_f32(
                false, At, false, Bt[t + 1], (short)0, cz1, false, false);
            float e0[8], e1[8];
#pragma unroll
            for (int r = 0; r < 8; ++r) e0[r] = __builtin_amdgcn_exp2f(d0[r]);
#pragma unroll
            for (int r = 0; r < 8; ++r) e1[r] = __builtin_amdgcn_exp2f(d1[r]);
#pragma unroll
            for (int r = 0; r < 8; ++r) esumA[r] += e0[r];
#pragma unroll
            for (int r = 0; r < 8; ++r) esumB[r] += e1[r];
        }

        // Merge streams, then sum over the 16 components held across each
        // 16-lane half.
#pragma unroll
        for (int r = 0; r < 8; ++r) {
            float v = esumA[r] + esumB[r];
            v += __shfl_xor(v, 1, 32);
            v += __shfl_xor(v, 2, 32);
            v += __shfl_xor(v, 4, 32);
            v += __shfl_xor(v, 8, 32);
            esumA[r] = v;
        }
        // Lane 0 owns rows 0-7 of this tile, lane 16 rows 8-15.
        // ln(S/256) = ln2 * (log2(S) - 8)
        if (ln == 0) {
            float s = 0.0f;
#pragma unroll
            for (int r = 0; r < 8; ++r)
                s += __builtin_amdgcn_logf(esumA[r]) - 8.0f;  // bare v_log_f32
            logqAcc += s * LN2_F;
        }
    }

    // Wave-level reduction (lanes >=16 carry zsq=0; only lanes 0/16 carry logq).
    float lq = logqAcc, zq = zsqAcc;
#pragma unroll
    for (int m = 16; m >= 1; m >>= 1) {
        lq += __shfl_xor(lq, m, 32);
        zq += __shfl_xor(zq, m, 32);
    }
    if (lane == 0) {
        atomicAdd(&logqSum[c * DIMZ + k], lq);
        atomicAdd(&zsqSum[c * DIMZ + k], zq);
    }
}

// ---------------------------------------------------------------------------
// Phase 3: okl = (1/nc) * sum_{c,k} [ logq[c,k] - logp[c,k] ]
// ---------------------------------------------------------------------------
__global__ void kld_finalize(const float* __restrict__ logqSum,
                             const float* __restrict__ zsqSum,
                             float* __restrict__ out) {
    __shared__ float red[NC * DIMZ];
    const int i = threadIdx.x;                 // 0..127 -> (c,k)
    const float inv = 1.0f / (float)SAMPLES;
    const float lq = logqSum[i] * inv;
    const float lp = -0.5f * (zsqSum[i] * inv + LOG2PI_F);
    red[i] = lq - lp;
    __syncthreads();
    for (int s = (NC * DIMZ) / 2; s >= 1; s >>= 1) {
        if (i < s) red[i] += red[i + s];
        __syncthreads();
    }
    if (i == 0) out[0] = red[0] / (float)NC;
}

extern "C" void kernel_launch(void* const* d_in, const int* in_sizes, int n_in,
                              void* d_out, int out_size, void* d_ws, size_t ws_size,
                              hipStream_t stream) {
    const float* mean   = (const float*)d_in[0];
    const float* logvar = (const float*)d_in[1];
    const float* eps    = (const float*)d_in[2];
    (void)in_sizes; (void)n_in; (void)out_size; (void)ws_size;

    float* ws      = (float*)d_ws;
    float* coefA   = ws + OFF_COEFA;
    float* coefB   = ws + OFF_COEFB;
    float* coefC   = ws + OFF_COEFC;
    float* stdv    = ws + OFF_STD;
    float* logqSum = ws + OFF_LOGQ;
    float* zsqSum  = ws + OFF_ZSQ;

    // Zero the atomic accumulators every launch (graph-capture safe).
    hipMemsetAsync(logqSum, 0, 2 * NC * DIMZ * sizeof(float), stream);

    kld_precomp<<<(BATCH * DIMZ + 255) / 256, 256, 0, stream>>>(
        mean, logvar, coefA, coefB, coefC, stdv);

    const int totalWaves = NC * DIMZ * GROUPS;       // 8192 waves
    kld_main<<<totalWaves / 8, 256, 0, stream>>>(
        mean, eps, coefA, coefB, coefC, stdv, logqSum, zsqSum);

    kld_finalize<<<1, NC * DIMZ, 0, stream>>>(logqSum, zsqSum, (float*)d_out);
}